// ChannelLatencySeq2Value_2920577762007
// MI455X (gfx1250) — compile-verified
//
#include <hip/hip_runtime.h>
#include <hip/hip_bf16.h>

// Problem constants (match reference)
#define BB   16
#define CC   64
#define TT   4096
#define DEC  128

// ALPHA = exp(-1/5) as f64 rounded to f32; BETA = (1 - ALPHA) in f64 rounded to f32
#define ALPHA_F 0.8187307530779818f
#define BETA_F  0.18126924692201815f

typedef float v2f __attribute__((ext_vector_type(2)));
typedef float v8f __attribute__((ext_vector_type(8)));

// ---------------------------------------------------------------------------
// Kernel A: encoder. Streams x once, writes dendritic_drive (B,C,4,T).
// Memory-bound: 16.7 MB read + 67 MB write; coalesced in t.
// ---------------------------------------------------------------------------
__global__ __launch_bounds__(256) void encode_kernel(
    const float* __restrict__ x,
    const float* __restrict__ w3, const float* __restrict__ b3,
    const float* __restrict__ w5, const float* __restrict__ b5,
    float* __restrict__ drive)
{
    int idx = blockIdx.x * 256 + threadIdx.x;        // < B*C*T = 4,194,304
    int t  = idx & (TT - 1);
    int bc = idx >> 12;                               // idx / T
    int c  = bc & (CC - 1);

    const float* X = x + (size_t)bc * TT;
    float xm2 = (t >= 2)      ? X[t - 2] : 0.0f;
    float xm1 = (t >= 1)      ? X[t - 1] : 0.0f;
    float x0  =                 X[t];
    float xp1 = (t + 1 < TT)  ? X[t + 1] : 0.0f;
    float xp2 = (t + 2 < TT)  ? X[t + 2] : 0.0f;

    const float* W3 = w3 + c * 6;    // (C*2,1,3) group-major
    const float* W5 = w5 + c * 10;   // (C*2,1,5)

    float o30 = W3[0]*xm1 + W3[1]*x0 + W3[2]*xp1 + b3[c*2 + 0];
    float o31 = W3[3]*xm1 + W3[4]*x0 + W3[5]*xp1 + b3[c*2 + 1];
    float o50 = W5[0]*xm2 + W5[1]*xm1 + W5[2]*x0 + W5[3]*xp1 + W5[4]*xp2 + b5[c*2 + 0];
    float o51 = W5[5]*xm2 + W5[6]*xm1 + W5[7]*x0 + W5[8]*xp1 + W5[9]*xp2 + b5[c*2 + 1];

    float* dr = drive + (size_t)bc * 4 * TT + t;
    dr[0]      = o30;
    dr[TT]     = o31;
    dr[2 * TT] = o50;
    dr[3 * TT] = o51;
}

// ---------------------------------------------------------------------------
// Kernel B: wave-parallel LIF first-crossing. One wave32 per (b,c) row.
// Affine-recurrence chunk scan: lane i owns t in [i*128, i*128+128).
//   pass1: chunk summary V_out = A128*V_in + S  (S = scan from 0)
//   shfl Kogge-Stone scan of affine maps -> exact V at chunk starts
//   pass2: rescan chunk from V_start, detect first crossing; min-reduce.
// ---------------------------------------------------------------------------
__global__ __launch_bounds__(256) void lif_kernel(
    const float* __restrict__ x,
    const float* __restrict__ w3, const float* __restrict__ b3,
    const float* __restrict__ w5, const float* __restrict__ b5,
    const float* __restrict__ rw, const float* __restrict__ rb,
    const float* __restrict__ latency_scale,
    float* __restrict__ true_lat, float* __restrict__ act)
{
    int gid  = blockIdx.x * 256 + threadIdx.x;
    int wave = gid >> 5;                // 0..1023 -> (b,c)
    int lane = gid & 31;
    int c    = wave & (CC - 1);

    const float* X = x + (size_t)wave * TT;

    // per-channel weights (wave-uniform -> scalar loads)
    const float* W3 = w3 + c * 6;
    const float* W5 = w5 + c * 10;
    float w30 = W3[0], w31 = W3[1], w32 = W3[2];
    float w33 = W3[3], w34 = W3[4], w35 = W3[5];
    float w50 = W5[0], w51 = W5[1], w52 = W5[2], w53 = W5[3], w54 = W5[4];
    float w55 = W5[5], w56 = W5[6], w57 = W5[7], w58 = W5[8], w59 = W5[9];
    float b30 = b3[c*2], b31 = b3[c*2 + 1];
    float b50 = b5[c*2], b51 = b5[c*2 + 1];
    float r0 = rw[c*4], r1 = rw[c*4 + 1], r2 = rw[c*4 + 2], r3 = rw[c*4 + 3];
    float rbc = rb[c];

    const int t0 = lane * 128;
    auto ldx = [&](int j) -> float { return (j >= 0 && j < TT) ? X[j] : 0.0f; };

    auto scan_chunk = [&](float V, bool detect, int& firstIdx) -> float {
        float xm2 = ldx(t0 - 2), xm1 = ldx(t0 - 1), x0 = ldx(t0);
        float xp1 = ldx(t0 + 1), xp2 = ldx(t0 + 2);
        firstIdx = 0x7fffffff;
        for (int k = 0; k < 128; ++k) {
            float o30 = w30*xm1 + w31*x0 + w32*xp1 + b30;
            float o31 = w33*xm1 + w34*x0 + w35*xp1 + b31;
            float o50 = w50*xm2 + w51*xm1 + w52*x0 + w53*xp1 + w54*xp2 + b50;
            float o51 = w55*xm2 + w56*xm1 + w57*x0 + w58*xp1 + w59*xp2 + b51;
            float d   = r0*o30 + r1*o31 + r2*o50 + r3*o51 + rbc;
            V = ALPHA_F * V + BETA_F * d;
            if (detect && firstIdx == 0x7fffffff && V >= 1.0f) firstIdx = t0 + k;
            xm2 = xm1; xm1 = x0; x0 = xp1; xp1 = xp2; xp2 = ldx(t0 + k + 3);
        }
        return V;
    };

    int dummy;
    float S = scan_chunk(0.0f, false, dummy);

    // inclusive scan of affine maps (A,B): next = cur o prev
    float A128 = __expf(-128.0f / 5.0f);   // alpha^128
    float Aacc = A128, Bacc = S;
    #pragma unroll
    for (int off = 1; off < 32; off <<= 1) {
        float Ap = __shfl_up(Aacc, off, 32);
        float Bp = __shfl_up(Bacc, off, 32);
        if (lane >= off) { Bacc = Aacc * Bp + Bacc; Aacc = Aacc * Ap; }
    }
    float Vstart = __shfl_up(Bacc, 1, 32);
    if (lane == 0) Vstart = 0.0f;

    int firstIdx;
    scan_chunk(Vstart, true, firstIdx);

    #pragma unroll
    for (int off = 16; off > 0; off >>= 1) {
        int o = __shfl_xor(firstIdx, off, 32);
        firstIdx = (o < firstIdx) ? o : firstIdx;
    }

    if (lane == 0) {
        float lat = (firstIdx == 0x7fffffff) ? (float)TT : (float)firstIdx;
        true_lat[wave] = lat;
        float scale = fmaxf(latency_scale[0], 0.001f);
        act[wave] = __expf(-lat / scale);
    }
}

// ---------------------------------------------------------------------------
// Kernel C: head MLP with V_WMMA_F32_16X16X4_F32 (exact f32).
// Single wave32; M=16 = batch. LDS stages act -> mixed -> h.
// Layouts per ISA 7.12.2 (32-bit A 16x4, 32-bit C/D 16x16).
// ---------------------------------------------------------------------------
__global__ __launch_bounds__(32) void mlp_head_kernel(
    const float* __restrict__ actv,   // (16,64)
    const float* __restrict__ gates,  // (64,64)  output_gates
    const float* __restrict__ w1,     // (128,64)
    const float* __restrict__ b1,     // (128)
    const float* __restrict__ w2,     // (64,128)
    const float* __restrict__ b2,     // (64)
    float* __restrict__ pred)         // (16,64)
{
    __shared__ float sA[16 * 64];
    __shared__ float sM[16 * 64];
    __shared__ float sH[16 * 128];

    const int lane = threadIdx.x;
    const int half = lane >> 4;       // 0/1
    const int lrow = lane & 15;

    for (int i = lane; i < 16 * 64; i += 32) sA[i] = actv[i];
    __syncthreads();

    // GEMM1: mixed(16,64) = act(16,64) x gates^T ; B[k][n] = gates[n][k]
    for (int n0 = 0; n0 < 64; n0 += 16) {
        v8f acc = {0, 0, 0, 0, 0, 0, 0, 0};
        for (int k0 = 0; k0 < 64; k0 += 4) {
            v2f a, bf;
            a.x  = sA[lrow * 64 + k0 + 2 * half];
            a.y  = sA[lrow * 64 + k0 + 2 * half + 1];
            bf.x = gates[(n0 + lrow) * 64 + k0 + 2 * half];
            bf.y = gates[(n0 + lrow) * 64 + k0 + 2 * half + 1];
            acc = __builtin_amdgcn_wmma_f32_16x16x4_f32(
                false, a, false, bf, (short)0, acc, false, false);
        }
        #pragma unroll
        for (int r = 0; r < 8; ++r)
            sM[(r + 8 * half) * 64 + n0 + lrow] = acc[r];
    }
    __syncthreads();

    // GEMM2: h(16,128) = relu(mixed x w1^T + b1)
    for (int n0 = 0; n0 < 128; n0 += 16) {
        v8f acc = {0, 0, 0, 0, 0, 0, 0, 0};
        for (int k0 = 0; k0 < 64; k0 += 4) {
            v2f a, bf;
            a.x  = sM[lrow * 64 + k0 + 2 * half];
            a.y  = sM[lrow * 64 + k0 + 2 * half + 1];
            bf.x = w1[(n0 + lrow) * 64 + k0 + 2 * half];
            bf.y = w1[(n0 + lrow) * 64 + k0 + 2 * half + 1];
            acc = __builtin_amdgcn_wmma_f32_16x16x4_f32(
                false, a, false, bf, (short)0, acc, false, false);
        }
        float bias = b1[n0 + lrow];
        #pragma unroll
        for (int r = 0; r < 8; ++r) {
            float v = acc[r] + bias;
            sH[(r + 8 * half) * 128 + n0 + lrow] = (v > 0.0f) ? v : 0.0f;
        }
    }
    __syncthreads();

    // GEMM3: raw(16,64) = h x w2^T + b2 ; pred = clip(softplus(raw), 0, T)
    for (int n0 = 0; n0 < 64; n0 += 16) {
        v8f acc = {0, 0, 0, 0, 0, 0, 0, 0};
        for (int k0 = 0; k0 < 128; k0 += 4) {
            v2f a, bf;
            a.x  = sH[lrow * 128 + k0 + 2 * half];
            a.y  = sH[lrow * 128 + k0 + 2 * half + 1];
            bf.x = w2[(n0 + lrow) * 128 + k0 + 2 * half];
            bf.y = w2[(n0 + lrow) * 128 + k0 + 2 * half + 1];
            acc = __builtin_amdgcn_wmma_f32_16x16x4_f32(
                false, a, false, bf, (short)0, acc, false, false);
        }
        float bias = b2[n0 + lrow];
        #pragma unroll
        for (int r = 0; r < 8; ++r) {
            float xv = acc[r] + bias;
            float sp = (xv > 20.0f) ? xv : log1pf(__expf(xv));
            float outv = fminf(fmaxf(sp, 0.0f), (float)TT);
            pred[(r + 8 * half) * 64 + n0 + lrow] = outv;
        }
    }
}

// ---------------------------------------------------------------------------
extern "C" void kernel_launch(void* const* d_in, const int* in_sizes, int n_in,
                              void* d_out, int out_size, void* d_ws, size_t ws_size,
                              hipStream_t stream) {
    const float* x   = (const float*)d_in[0];
    const float* w3  = (const float*)d_in[1];
    const float* b3  = (const float*)d_in[2];
    const float* w5  = (const float*)d_in[3];
    const float* b5  = (const float*)d_in[4];
    const float* rw  = (const float*)d_in[5];
    const float* rb  = (const float*)d_in[6];
    const float* ls  = (const float*)d_in[7];
    const float* og  = (const float*)d_in[8];
    const float* mw1 = (const float*)d_in[9];
    const float* mb1 = (const float*)d_in[10];
    const float* mw2 = (const float*)d_in[11];
    const float* mb2 = (const float*)d_in[12];

    float* out   = (float*)d_out;
    float* pred  = out;                       // (16,64)
    float* tlat  = out + BB * CC;             // (16,64)
    float* act   = out + 2 * BB * CC;         // (16,64)
    float* drive = out + 3 * BB * CC;         // (16,64,4,4096)

    encode_kernel<<<(BB * CC * TT) / 256, 256, 0, stream>>>(x, w3, b3, w5, b5, drive);
    lif_kernel<<<(BB * CC * 32) / 256, 256, 0, stream>>>(x, w3, b3, w5, b5, rw, rb, ls,
                                                         tlat, act);
    mlp_head_kernel<<<1, 32, 0, stream>>>(act, og, mw1, mb1, mw2, mb2, pred);
}